// RecurrentNet_70583492542687
// MI455X (gfx1250) — compile-verified
//
#include <hip/hip_runtime.h>
#include <hip/hip_bf16.h>
#include <math.h>

typedef __attribute__((ext_vector_type(2))) float v2f;
typedef __attribute__((ext_vector_type(8))) float v8f;

#define T_SEQ 32
#define B_SZ  64
#define E_DIM 512
#define D_DIM 512
#define S_CTX 128
#define H_DIM 1024
#define V_DIM 32000
#define IN_DIM 1026   // E + D + 2

// ---------------------------------------------------------------------------
// Utility: zero a buffer (d_ws / d_out regions are poisoned by the harness)
// ---------------------------------------------------------------------------
__global__ void zero_kernel(float* __restrict__ p, int n) {
    int i = blockIdx.x * blockDim.x + threadIdx.x;
    if (i < n) p[i] = 0.0f;
}

// ---------------------------------------------------------------------------
// Fused attention step: q = h@Wq, scores = q . convfeat, softmax, ctx,
// build GRU input [x_t | ctx | pos1 | pos2], accumulate ctx into ctx_sum.
// One block (256 threads) per batch element.
// ---------------------------------------------------------------------------
__global__ __launch_bounds__(256) void attn_step_kernel(
    const float* __restrict__ h,        // [B][H]
    const float* __restrict__ Wq,       // [H][D] row-major
    const float* __restrict__ cf,       // [B][D][S]
    const float* __restrict__ xt,       // [B][E]  (= seqembd[t])
    const float* __restrict__ pos1,     // [B]
    const float* __restrict__ pos2,     // [B]
    float* __restrict__ inp,            // [B][IN_DIM]
    float* __restrict__ attn_out,       // [B][S]  (slice of d_out)
    float* __restrict__ ctx_sum)        // [B][D]  (slice of d_out, accumulated)
{
    const int b   = blockIdx.x;
    const int tid = threadIdx.x;

    __shared__ float sh_h[H_DIM];
    __shared__ float sh_q[D_DIM];
    __shared__ float sh_attn[S_CTX];
    __shared__ float red[S_CTX];

    const float* hb = h + (size_t)b * H_DIM;
    for (int i = tid; i < H_DIM; i += 256) sh_h[i] = hb[i];
    __syncthreads();

    // q = h @ Wq : each thread computes 2 outputs, Wq reads coalesced over d
    float q0 = 0.0f, q1 = 0.0f;
    for (int k = 0; k < H_DIM; ++k) {
        const float hv = sh_h[k];
        const float* wr = Wq + (size_t)k * D_DIM;
        q0 = fmaf(hv, wr[tid], q0);
        q1 = fmaf(hv, wr[tid + 256], q1);
    }
    sh_q[tid] = q0;
    sh_q[tid + 256] = q1;
    __syncthreads();

    // scores[s] = sum_d q[d] * cf[b][d][s]   (coalesced over s)
    const float* cfb = cf + (size_t)b * D_DIM * S_CTX;
    float sc = 0.0f;
    if (tid < S_CTX) {
        for (int d = 0; d < D_DIM; ++d)
            sc = fmaf(sh_q[d], cfb[(size_t)d * S_CTX + tid], sc);
        red[tid] = sc;
    }
    __syncthreads();
    for (int off = 64; off > 0; off >>= 1) {
        if (tid < off) red[tid] = fmaxf(red[tid], red[tid + off]);
        __syncthreads();
    }
    const float mx = red[0];
    __syncthreads();
    const float ex = (tid < S_CTX) ? __expf(sc - mx) : 0.0f;
    if (tid < S_CTX) red[tid] = ex;
    __syncthreads();
    for (int off = 64; off > 0; off >>= 1) {
        if (tid < off) red[tid] += red[tid + off];
        __syncthreads();
    }
    const float inv = 1.0f / red[0];
    if (tid < S_CTX) {
        const float a = ex * inv;
        sh_attn[tid] = a;
        attn_out[(size_t)b * S_CTX + tid] = a;
    }
    __syncthreads();

    // ctx[d] = sum_s attn[s] * cf[b][d][s]; assemble GRU input row
    float* inpb = inp + (size_t)b * IN_DIM;
    for (int d = tid; d < D_DIM; d += 256) {
        const float* row = cfb + (size_t)d * S_CTX;
        float c = 0.0f;
        for (int s = 0; s < S_CTX; ++s) c = fmaf(sh_attn[s], row[s], c);
        inpb[E_DIM + d] = c;                       // ctx part
        ctx_sum[(size_t)b * D_DIM + d] += c;       // running sum over t
        inpb[d] = xt[(size_t)b * E_DIM + d];       // x_t part (E==D==512)
    }
    if (tid == 0) {
        inpb[E_DIM + D_DIM]     = pos1[b];
        inpb[E_DIM + D_DIM + 1] = pos2[b];
    }
}

// ---------------------------------------------------------------------------
// FP32 WMMA GEMM:  C[M][N] = A[M][K] * B[N][K]^T (+ bias[N])
// A row-major (lda), B row-major K-contiguous (ldb) -> both fragments are
// contiguous float pairs per lane (merge to global_load_b64).
// Uses V_WMMA_F32_16X16X4_F32. Block = 256 threads (8 waves).
// Each wave: 2 M-tiles x 4 N-tiles (A frag reused 4x, B frag reused 2x).
// grid.y = M/32, grid.x = ceil(N/512); every live wave covers a full 64-col
// group (launches below tile N exactly), so the only guard is a SCALAR
// early-out (readfirstlane) -> EXEC is provably all-ones around every WMMA,
// as the ISA requires. Masked tail (per-lane v_cndmask, no EXEC change)
// handles K % 4 != 0.
// ---------------------------------------------------------------------------
#define WMMA_F32(a, b, c) \
    __builtin_amdgcn_wmma_f32_16x16x4_f32(false, (a), false, (b), (short)0, (c), false, false)

__global__ __launch_bounds__(256) void wmma_gemm_f32_nt(
    const float* __restrict__ A, const float* __restrict__ Bm,
    const float* __restrict__ bias, float* __restrict__ C,
    int N, int K, int lda, int ldb, int ldc)
{
    const int lane = threadIdx.x & 31;
    // Wave index as an SGPR value: makes the bounds check a scalar branch
    // (s_cmp/s_cbranch) instead of exec-mask manipulation.
    const int wave = __builtin_amdgcn_readfirstlane(threadIdx.x >> 5);
    const int lm   = lane & 15;        // row/col within tile
    const int kh   = lane >> 4;        // K-half select (0/1)
    const int ko   = kh * 2;           // K offset within 4-step

    const int m0 = blockIdx.y * 32;
    const int n0 = (blockIdx.x * 8 + wave) * 64;
    if (n0 + 64 > N) return;           // scalar, wave-uniform

    const float* A0 = A  + (size_t)(m0 + lm)      * lda;
    const float* A1 = A  + (size_t)(m0 + 16 + lm) * lda;
    const float* B0 = Bm + (size_t)(n0 + lm)      * ldb;
    const float* B1 = Bm + (size_t)(n0 + 16 + lm) * ldb;
    const float* B2 = Bm + (size_t)(n0 + 32 + lm) * ldb;
    const float* B3 = Bm + (size_t)(n0 + 48 + lm) * ldb;

    const v8f z = {0.f, 0.f, 0.f, 0.f, 0.f, 0.f, 0.f, 0.f};
    v8f acc00 = z, acc01 = z, acc02 = z, acc03 = z;
    v8f acc10 = z, acc11 = z, acc12 = z, acc13 = z;

    int k = 0;
    for (; k + 4 <= K; k += 4) {
        v2f a0, a1, b0, b1, b2, b3;
        a0[0] = A0[k + ko]; a0[1] = A0[k + ko + 1];
        a1[0] = A1[k + ko]; a1[1] = A1[k + ko + 1];
        b0[0] = B0[k + ko]; b0[1] = B0[k + ko + 1];
        b1[0] = B1[k + ko]; b1[1] = B1[k + ko + 1];
        b2[0] = B2[k + ko]; b2[1] = B2[k + ko + 1];
        b3[0] = B3[k + ko]; b3[1] = B3[k + ko + 1];

        acc00 = WMMA_F32(a0, b0, acc00);
        acc10 = WMMA_F32(a1, b0, acc10);
        acc01 = WMMA_F32(a0, b1, acc01);
        acc11 = WMMA_F32(a1, b1, acc11);
        acc02 = WMMA_F32(a0, b2, acc02);
        acc12 = WMMA_F32(a1, b2, acc12);
        acc03 = WMMA_F32(a0, b3, acc03);
        acc13 = WMMA_F32(a1, b3, acc13);
    }
    if (k < K) {  // masked tail (per-lane selects, no EXEC divergence)
        const bool e0 = (k + ko)     < K;
        const bool e1 = (k + ko + 1) < K;
        v2f a0, a1, b0, b1, b2, b3;
        a0[0] = e0 ? A0[k + ko] : 0.f; a0[1] = e1 ? A0[k + ko + 1] : 0.f;
        a1[0] = e0 ? A1[k + ko] : 0.f; a1[1] = e1 ? A1[k + ko + 1] : 0.f;
        b0[0] = e0 ? B0[k + ko] : 0.f; b0[1] = e1 ? B0[k + ko + 1] : 0.f;
        b1[0] = e0 ? B1[k + ko] : 0.f; b1[1] = e1 ? B1[k + ko + 1] : 0.f;
        b2[0] = e0 ? B2[k + ko] : 0.f; b2[1] = e1 ? B2[k + ko + 1] : 0.f;
        b3[0] = e0 ? B3[k + ko] : 0.f; b3[1] = e1 ? B3[k + ko + 1] : 0.f;

        acc00 = WMMA_F32(a0, b0, acc00);
        acc10 = WMMA_F32(a1, b0, acc10);
        acc01 = WMMA_F32(a0, b1, acc01);
        acc11 = WMMA_F32(a1, b1, acc11);
        acc02 = WMMA_F32(a0, b2, acc02);
        acc12 = WMMA_F32(a1, b2, acc12);
        acc03 = WMMA_F32(a0, b3, acc03);
        acc13 = WMMA_F32(a1, b3, acc13);
    }

    // Store. C/D layout: VGPR i holds row m0 + i + 8*kh, col = lm.
    const int rbase = m0 + kh * 8;
    {
        const int col = n0 + lm;
        const float bv = bias ? bias[col] : 0.0f;
        #pragma unroll
        for (int i = 0; i < 8; ++i) {
            C[(size_t)(rbase + i)      * ldc + col] = acc00[i] + bv;
            C[(size_t)(rbase + 16 + i) * ldc + col] = acc10[i] + bv;
        }
    }
    {
        const int col = n0 + 16 + lm;
        const float bv = bias ? bias[col] : 0.0f;
        #pragma unroll
        for (int i = 0; i < 8; ++i) {
            C[(size_t)(rbase + i)      * ldc + col] = acc01[i] + bv;
            C[(size_t)(rbase + 16 + i) * ldc + col] = acc11[i] + bv;
        }
    }
    {
        const int col = n0 + 32 + lm;
        const float bv = bias ? bias[col] : 0.0f;
        #pragma unroll
        for (int i = 0; i < 8; ++i) {
            C[(size_t)(rbase + i)      * ldc + col] = acc02[i] + bv;
            C[(size_t)(rbase + 16 + i) * ldc + col] = acc12[i] + bv;
        }
    }
    {
        const int col = n0 + 48 + lm;
        const float bv = bias ? bias[col] : 0.0f;
        #pragma unroll
        for (int i = 0; i < 8; ++i) {
            C[(size_t)(rbase + i)      * ldc + col] = acc03[i] + bv;
            C[(size_t)(rbase + 16 + i) * ldc + col] = acc13[i] + bv;
        }
    }
}

// ---------------------------------------------------------------------------
// GRU elementwise (torch semantics). In-place h update is safe: h enters only
// elementwise as z*h.
// ---------------------------------------------------------------------------
__global__ void gru_kernel(const float* __restrict__ gi, const float* __restrict__ gh,
                           const float* __restrict__ b_ih, const float* __restrict__ b_hh,
                           float* __restrict__ h)
{
    int idx = blockIdx.x * blockDim.x + threadIdx.x;
    if (idx >= B_SZ * H_DIM) return;
    const int b = idx / H_DIM;
    const int j = idx - b * H_DIM;
    const float* gib = gi + (size_t)b * 3 * H_DIM;
    const float* ghb = gh + (size_t)b * 3 * H_DIM;

    const float ir = gib[j]             + b_ih[j];
    const float iz = gib[H_DIM + j]     + b_ih[H_DIM + j];
    const float in = gib[2 * H_DIM + j] + b_ih[2 * H_DIM + j];
    const float hr = ghb[j]             + b_hh[j];
    const float hz = ghb[H_DIM + j]     + b_hh[H_DIM + j];
    const float hn = ghb[2 * H_DIM + j] + b_hh[2 * H_DIM + j];

    const float r = 1.0f / (1.0f + __expf(-(ir + hr)));
    const float zg = 1.0f / (1.0f + __expf(-(iz + hz)));
    const float n = tanhf(in + r * hn);
    h[idx] = (1.0f - zg) * n + zg * h[idx];
}

// ---------------------------------------------------------------------------
// Per-row max + log-sum-exp over V=32000 (one block per batch row; logits
// are L2-resident, two passes).
// ---------------------------------------------------------------------------
__global__ __launch_bounds__(256) void rowstats_kernel(
    const float* __restrict__ logits, float* __restrict__ stats)
{
    const int b = blockIdx.x;
    const int tid = threadIdx.x;
    const float* row = logits + (size_t)b * V_DIM;
    __shared__ float red[256];

    float m = -INFINITY;
    for (int v = tid; v < V_DIM; v += 256) m = fmaxf(m, row[v]);
    red[tid] = m;
    __syncthreads();
    for (int off = 128; off > 0; off >>= 1) {
        if (tid < off) red[tid] = fmaxf(red[tid], red[tid + off]);
        __syncthreads();
    }
    m = red[0];
    __syncthreads();

    float s = 0.0f;
    for (int v = tid; v < V_DIM; v += 256) s += __expf(row[v] - m);
    red[tid] = s;
    __syncthreads();
    for (int off = 128; off > 0; off >>= 1) {
        if (tid < off) red[tid] += red[tid + off];
        __syncthreads();
    }
    if (tid == 0) {
        stats[b * 2]     = m;
        stats[b * 2 + 1] = __logf(red[0]);
    }
}

__global__ void write_lp_kernel(const float* __restrict__ logits,
                                const float* __restrict__ stats,
                                float* __restrict__ out)
{
    int idx = blockIdx.x * blockDim.x + threadIdx.x;
    if (idx >= B_SZ * V_DIM) return;
    const int b = idx / V_DIM;
    out[idx] = logits[idx] - stats[b * 2] - stats[b * 2 + 1];
}

// ---------------------------------------------------------------------------
// Launch
// ---------------------------------------------------------------------------
extern "C" void kernel_launch(void* const* d_in, const int* in_sizes, int n_in,
                              void* d_out, int out_size, void* d_ws, size_t ws_size,
                              hipStream_t stream)
{
    (void)in_sizes; (void)n_in; (void)out_size; (void)ws_size;
    const float* seqembd = (const float*)d_in[0];
    /* d_in[1] = lengths (int), unused by reference forward */
    const float* cf    = (const float*)d_in[2];
    const float* pos1  = (const float*)d_in[3];
    const float* pos2  = (const float*)d_in[4];
    const float* Wq    = (const float*)d_in[5];
    const float* w_ih  = (const float*)d_in[6];
    const float* w_hh  = (const float*)d_in[7];
    const float* b_ih  = (const float*)d_in[8];
    const float* b_hh  = (const float*)d_in[9];
    const float* W_out = (const float*)d_in[10];
    const float* b_out = (const float*)d_in[11];

    float* out = (float*)d_out;
    const size_t OFF_CTX = (size_t)T_SEQ * B_SZ * V_DIM;            // contexts sum
    const size_t OFF_H   = OFF_CTX + (size_t)B_SZ * D_DIM;          // h_last
    const size_t OFF_AT  = OFF_H + (size_t)B_SZ * H_DIM;            // attentions

    float* ws     = (float*)d_ws;
    float* h      = ws;                                   // B*H      = 65536
    float* inp    = h      + (size_t)B_SZ * H_DIM;        // B*IN     = 65664
    float* gi     = inp    + (size_t)B_SZ * IN_DIM;       // B*3H     = 196608
    float* gh     = gi     + (size_t)B_SZ * 3 * H_DIM;    // B*3H     = 196608
    float* logits = gh     + (size_t)B_SZ * 3 * H_DIM;    // B*V      = 2048000
    float* stats  = logits + (size_t)B_SZ * V_DIM;        // 2*B      = 128

    zero_kernel<<<(B_SZ * H_DIM + 255) / 256, 256, 0, stream>>>(h, B_SZ * H_DIM);
    zero_kernel<<<(B_SZ * D_DIM + 255) / 256, 256, 0, stream>>>(out + OFF_CTX, B_SZ * D_DIM);

    const dim3 gemm_blk(256, 1, 1);
    const dim3 g_gates((3 * H_DIM + 511) / 512, B_SZ / 32, 1);  // (6, 2)
    const dim3 g_logit((V_DIM + 511) / 512,     B_SZ / 32, 1);  // (63, 2)

    for (int t = 0; t < T_SEQ; ++t) {
        attn_step_kernel<<<B_SZ, 256, 0, stream>>>(
            h, Wq, cf, seqembd + (size_t)t * B_SZ * E_DIM, pos1, pos2,
            inp, out + OFF_AT + (size_t)t * B_SZ * S_CTX, out + OFF_CTX);

        wmma_gemm_f32_nt<<<g_gates, gemm_blk, 0, stream>>>(
            inp, w_ih, nullptr, gi, 3 * H_DIM, IN_DIM, IN_DIM, IN_DIM, 3 * H_DIM);
        wmma_gemm_f32_nt<<<g_gates, gemm_blk, 0, stream>>>(
            h, w_hh, nullptr, gh, 3 * H_DIM, H_DIM, H_DIM, H_DIM, 3 * H_DIM);

        gru_kernel<<<(B_SZ * H_DIM + 255) / 256, 256, 0, stream>>>(gi, gh, b_ih, b_hh, h);

        wmma_gemm_f32_nt<<<g_logit, gemm_blk, 0, stream>>>(
            h, W_out, b_out, logits, V_DIM, H_DIM, H_DIM, H_DIM, V_DIM);

        rowstats_kernel<<<B_SZ, 256, 0, stream>>>(logits, stats);
        write_lp_kernel<<<(B_SZ * V_DIM + 255) / 256, 256, 0, stream>>>(
            logits, stats, out + (size_t)t * B_SZ * V_DIM);
    }

    hipMemcpyAsync(out + OFF_H, h, (size_t)B_SZ * H_DIM * sizeof(float),
                   hipMemcpyDeviceToDevice, stream);
}